// SSMLayer_68247030333813
// MI455X (gfx1250) — compile-verified
//
#include <hip/hip_runtime.h>
#include <math.h>

// SSM layer: B=4, L=4096, D=1024, S=16
#define BATCH  4
#define SEQ    4096
#define DIM    1024
#define STATE  16
#define BL     (BATCH * SEQ)          // 16384 fused (b,l) rows
#define CH     64                     // scan chunk length
#define NCHUNK (BL / CH)              // 256 chunks (64 per batch; never crosses batch)

typedef __attribute__((ext_vector_type(2))) float v2f;
typedef __attribute__((ext_vector_type(4))) float v4f;
typedef __attribute__((ext_vector_type(8))) float v8f;

// D = A(16x4, f32) * B(4x16, f32) + C(16x16, f32)  -- V_WMMA_F32_16X16X4_F32
__device__ __forceinline__ v8f wmma4(v2f a, v2f b, v8f c) {
    return __builtin_amdgcn_wmma_f32_16x16x4_f32(
        /*neg_a=*/false, a, /*neg_b=*/false, b,
        /*c_mod=*/(short)0, c, /*reuse_a=*/false, /*reuse_b=*/false);
}

// ---------------------------------------------------------------- K0: A^{-1}
__global__ __launch_bounds__(256)
void inv16_kernel(const float* __restrict__ A, float* __restrict__ Ainv) {
    __shared__ float M[16][16];
    __shared__ float Inv[16][16];
    __shared__ float fac[16];
    int j = threadIdx.x & 15, i = threadIdx.x >> 4;
    M[i][j]   = A[i * 16 + j];
    Inv[i][j] = (i == j) ? 1.f : 0.f;
    __syncthreads();
    for (int k = 0; k < 16; ++k) {
        float piv = M[k][k];                 // everyone reads before row-k scale
        __syncthreads();
        if (i == k) { M[k][j] /= piv; Inv[k][j] /= piv; }
        __syncthreads();
        if (j == 0) fac[i] = M[i][k];
        __syncthreads();
        if (i != k) {
            float f = fac[i];
            M[i][j]   -= f * M[k][j];
            Inv[i][j] -= f * Inv[k][j];
        }
        __syncthreads();
    }
    Ainv[i * 16 + j] = Inv[i][j];
}

// ------------------------- K1: projections + softplus + dA/dB/C per 16 rows
__global__ __launch_bounds__(32)
void proj_kernel(const float* __restrict__ x,
                 const float* __restrict__ Wb, const float* __restrict__ bb,
                 const float* __restrict__ Wc, const float* __restrict__ bc,
                 const float* __restrict__ Wd, const float* __restrict__ bd,
                 const float* __restrict__ A,  const float* __restrict__ Ainv,
                 float* __restrict__ dAout, float* __restrict__ dBout,
                 float* __restrict__ Cout) {
    const int l0   = blockIdx.x * 16;        // tile of 16 fused rows
    const int lane = threadIdx.x;
    const int half = lane >> 4;              // K-split lo/hi per ISA A/B layout
    const int ln   = lane & 15;

    __shared__ float sA[256], sAinv[256];
    __shared__ float sDelta[16][16];         // [m][state]
    __shared__ float sU[16][16];             // delta * B
    __shared__ float sW[16][16];             // Ainv @ (delta*B)
#pragma unroll
    for (int r = 0; r < 8; ++r) {
        int idx = r * 32 + lane;
        sA[idx] = A[idx]; sAinv[idx] = Ainv[idx];
    }

    const float* xrow = x + (size_t)(l0 + ln) * DIM;   // A-frag row M=ln
    const float* wbr  = Wb + (size_t)ln * DIM;         // B-frag col N=ln (W^T row)
    const float* wcr  = Wc + (size_t)ln * DIM;
    const float* wdr  = Wd + (size_t)ln * DIM;

    v8f accB = {}, accC = {}, accD = {};
    for (int k0 = 0; k0 < DIM; k0 += 4) {
        const int k = k0 + half * 2;
        if ((k0 & 63) == 0) __builtin_prefetch(xrow + k + 128, 0, 1);
        v2f a  = *(const v2f*)(xrow + k);
        accB = wmma4(a, *(const v2f*)(wbr + k), accB);
        accC = wmma4(a, *(const v2f*)(wcr + k), accC);
        accD = wmma4(a, *(const v2f*)(wdr + k), accD);
    }

    const float bbn = bb[ln], bcn = bc[ln], bdn = bd[ln];
#pragma unroll
    for (int r = 0; r < 8; ++r) {            // D-frag: (m = r + 8*half, n = ln)
        int m = r + 8 * half;
        float Bv = accB[r] + bbn;
        float Cv = accC[r] + bcn;
        float z  = accD[r] + bdn;
        float dl = (z > 20.f) ? z : log1pf(expf(z));   // softplus
        sDelta[m][ln] = dl;
        sU[m][ln]     = dl * Bv;
        Cout[(size_t)(l0 + m) * STATE + ln] = Cv;
    }
    __syncthreads();

    // w[m][j] = sum_k Ainv[j][k] * u[m][k]
#pragma unroll
    for (int r = 0; r < 8; ++r) {
        int idx = r * 32 + lane, m = idx >> 4, j = idx & 15;
        float s = 0.f;
#pragma unroll
        for (int k = 0; k < 16; ++k) s += sAinv[j * 16 + k] * sU[m][k];
        sW[m][j] = s;
    }
    __syncthreads();

    // dA[m][i][j] = exp(delta[m][i] * A[i][j]);  dB[m][i] = sum_j dA*w - w[i]
#pragma unroll
    for (int mm = 0; mm < 8; ++mm) {
        int m = mm * 2 + half;               // both half-waves busy
        int i = ln;
        float di  = sDelta[m][i];
        float acc = 0.f;
        float row[16];
#pragma unroll
        for (int j = 0; j < 16; ++j) {
            float da = expf(di * sA[i * 16 + j]);
            row[j] = da;
            acc += da * sW[m][j];
        }
        float* dst = dAout + (size_t)(l0 + m) * 256 + i * 16;
        *(v4f*)(dst + 0)  = *(v4f*)(row + 0);
        *(v4f*)(dst + 4)  = *(v4f*)(row + 4);
        *(v4f*)(dst + 8)  = *(v4f*)(row + 8);
        *(v4f*)(dst + 12) = *(v4f*)(row + 12);
        dBout[(size_t)(l0 + m) * STATE + i] = acc - sW[m][i];
    }
}

// --------------- K2: chunk summaries  P_c = prod dA_t (WMMA),  q_c = scan(0)
__global__ __launch_bounds__(32)
void chunk_kernel(const float* __restrict__ dA, const float* __restrict__ dB,
                  float* __restrict__ Pc, float* __restrict__ qc) {
    const int chunk = blockIdx.x, t0 = chunk * CH;
    const int lane = threadIdx.x, half = lane >> 4, ln = lane & 15;
    __shared__ float sP[16][16];
    __shared__ float sq[16];
#pragma unroll
    for (int r = 0; r < 8; ++r) {
        int idx = r * 32 + lane;
        sP[idx >> 4][idx & 15] = ((idx >> 4) == (idx & 15)) ? 1.f : 0.f;
    }
    if (lane < 16) sq[lane] = 0.f;
    __syncthreads();

    for (int t = t0; t < t0 + CH; ++t) {
        const float* M = dA + (size_t)t * 256;
        // Pn = dA_t @ P   (A-frag = dA rows, B-frag = P columns from LDS)
        v8f acc = {};
#pragma unroll
        for (int k0 = 0; k0 < 16; k0 += 4) {
            int k = k0 + half * 2;
            v2f a = *(const v2f*)(M + ln * 16 + k);
            v2f b; b.x = sP[k][ln]; b.y = sP[k + 1][ln];
            acc = wmma4(a, b, acc);
        }
        // qn = dA_t @ q + dB_t
        float qn = 0.f;
        if (lane < 16) {
#pragma unroll
            for (int j = 0; j < 16; ++j) qn += M[ln * 16 + j] * sq[j];
            qn += dB[(size_t)t * STATE + ln];
        }
        __syncthreads();
#pragma unroll
        for (int r = 0; r < 8; ++r) sP[r + 8 * half][ln] = acc[r];
        if (lane < 16) sq[ln] = qn;
        __syncthreads();
    }
#pragma unroll
    for (int r = 0; r < 8; ++r) {
        int m = r + 8 * half;
        Pc[(size_t)chunk * 256 + m * 16 + ln] = sP[m][ln];
    }
    if (lane < 16) qc[(size_t)chunk * STATE + ln] = sq[ln];
}

// ---------------------- K3: per-batch boundary scan over 64 chunk summaries
__global__ __launch_bounds__(32)
void boundary_kernel(const float* __restrict__ Pc, const float* __restrict__ qc,
                     float* __restrict__ hin) {
    const int b = blockIdx.x, lane = threadIdx.x;
    const int cpb = SEQ / CH;                // 64 chunks per batch
    __shared__ float sh[16];
    if (lane < 16) sh[lane] = 0.f;
    __syncthreads();
    for (int c = 0; c < cpb; ++c) {
        int chunk = b * cpb + c;
        if (lane < 16) hin[(size_t)chunk * STATE + lane] = sh[lane];
        __syncthreads();
        float hn = 0.f;
        if (lane < 16) {
            const float* P = Pc + (size_t)chunk * 256 + lane * 16;
#pragma unroll
            for (int j = 0; j < 16; ++j) hn += P[j] * sh[j];
            hn += qc[(size_t)chunk * STATE + lane];
        }
        __syncthreads();
        if (lane < 16) sh[lane] = hn;
        __syncthreads();
    }
}

// ----------------------- K4: within-chunk replay, emit y_t = C_t (*) h_t
__global__ __launch_bounds__(32)
void scan_kernel(const float* __restrict__ dA, const float* __restrict__ dB,
                 const float* __restrict__ Cp, const float* __restrict__ hin,
                 float* __restrict__ yv) {
    const int chunk = blockIdx.x, t0 = chunk * CH;
    const int lane = threadIdx.x;
    __shared__ float sh[16];
    if (lane < 16) sh[lane] = hin[(size_t)chunk * STATE + lane];
    __syncthreads();
    for (int t = t0; t < t0 + CH; ++t) {
        float hn = 0.f;
        if (lane < 16) {
            const float* M = dA + (size_t)t * 256 + lane * 16;
            v4f m0 = *(const v4f*)(M + 0),  m1 = *(const v4f*)(M + 4);
            v4f m2 = *(const v4f*)(M + 8),  m3 = *(const v4f*)(M + 12);
            hn = m0.x*sh[0] + m0.y*sh[1] + m0.z*sh[2] + m0.w*sh[3]
               + m1.x*sh[4] + m1.y*sh[5] + m1.z*sh[6] + m1.w*sh[7]
               + m2.x*sh[8] + m2.y*sh[9] + m2.z*sh[10]+ m2.w*sh[11]
               + m3.x*sh[12]+ m3.y*sh[13]+ m3.z*sh[14]+ m3.w*sh[15];
            hn += dB[(size_t)t * STATE + lane];
        }
        __syncthreads();
        if (lane < 16) {
            sh[lane] = hn;
            yv[(size_t)t * STATE + lane] = Cp[(size_t)t * STATE + lane] * hn;
        }
        __syncthreads();
    }
}

// ---------------------------------- K5: out = y @ Wout^T + bout  (K=16 WMMA)
__global__ __launch_bounds__(32)
void out_kernel(const float* __restrict__ yv, const float* __restrict__ Wout,
                const float* __restrict__ bout, float* __restrict__ out) {
    const int tile  = blockIdx.x;
    const int nT    = tile & (DIM / 16 - 1);
    const int mT    = tile >> 6;             // DIM/16 == 64
    const int l0 = mT * 16, n0 = nT * 16;
    const int lane = threadIdx.x, half = lane >> 4, ln = lane & 15;

    v8f acc = {};
#pragma unroll
    for (int k0 = 0; k0 < STATE; k0 += 4) {
        int k = k0 + half * 2;
        v2f a = *(const v2f*)(yv + (size_t)(l0 + ln) * STATE + k);
        v2f b = *(const v2f*)(Wout + (size_t)(n0 + ln) * STATE + k);
        acc = wmma4(a, b, acc);
    }
    const float bn = bout[n0 + ln];
#pragma unroll
    for (int r = 0; r < 8; ++r) {
        int m = r + 8 * half;
        out[(size_t)(l0 + m) * DIM + n0 + ln] = acc[r] + bn;
    }
}

extern "C" void kernel_launch(void* const* d_in, const int* in_sizes, int n_in,
                              void* d_out, int out_size, void* d_ws, size_t ws_size,
                              hipStream_t stream) {
    (void)in_sizes; (void)n_in; (void)out_size; (void)ws_size;
    const float* x    = (const float*)d_in[0];
    const float* Wb   = (const float*)d_in[1];
    const float* bb   = (const float*)d_in[2];
    const float* Wc   = (const float*)d_in[3];
    const float* bc   = (const float*)d_in[4];
    const float* Wd   = (const float*)d_in[5];
    const float* bd   = (const float*)d_in[6];
    const float* A    = (const float*)d_in[7];
    const float* Wout = (const float*)d_in[8];
    const float* bout = (const float*)d_in[9];
    float* out = (float*)d_out;

    float* ws   = (float*)d_ws;
    float* Ainv = ws;                                  // 256
    float* dAw  = Ainv + 256;                          // BL*256 = 4.19M floats
    float* dBw  = dAw + (size_t)BL * 256;              // BL*16
    float* Cw   = dBw + (size_t)BL * STATE;            // BL*16
    float* yw   = Cw  + (size_t)BL * STATE;            // BL*16
    float* Pcw  = yw  + (size_t)BL * STATE;            // NCHUNK*256
    float* qcw  = Pcw + (size_t)NCHUNK * 256;          // NCHUNK*16
    float* hinw = qcw + (size_t)NCHUNK * STATE;        // NCHUNK*16

    inv16_kernel  <<<1, 256, 0, stream>>>(A, Ainv);
    proj_kernel   <<<BL / 16, 32, 0, stream>>>(x, Wb, bb, Wc, bc, Wd, bd, A, Ainv,
                                               dAw, dBw, Cw);
    chunk_kernel  <<<NCHUNK, 32, 0, stream>>>(dAw, dBw, Pcw, qcw);
    boundary_kernel<<<BATCH, 32, 0, stream>>>(Pcw, qcw, hinw);
    scan_kernel   <<<NCHUNK, 32, 0, stream>>>(dAw, dBw, Cw, hinw, yw);
    out_kernel    <<<(BL / 16) * (DIM / 16), 32, 0, stream>>>(yw, Wout, bout, out);
}